// AGSM_57655640981780
// MI455X (gfx1250) — compile-verified
//
#include <hip/hip_runtime.h>
#include <hip/hip_bf16.h>

// ---------------- problem constants ----------------
#define N_IMG   128         // T*16
#define C_CH    64
#define H_DIM   56
#define W_DIM   56
#define T_DIM   8
#define B_DIM   16          // N/T
#define HW      3136        // H*W
#define CHW     200704      // C*H*W
#define TOTAL   25690112    // N*C*H*W
#define EPS_BN  1e-5f

// ---------------- workspace layout (float offsets) ----------------
#define X2_OFF    0
#define GATE_OFF  25690112                  // B*2*T*H*W = 802816 floats
#define SUM_OFF   (GATE_OFF + 802816)
#define SQ_OFF    (SUM_OFF + 64)
#define SCALE_OFF (SQ_OFF + 64)
#define SHIFT_OFF (SCALE_OFF + 64)

typedef __attribute__((ext_vector_type(2))) float v2f;
typedef __attribute__((ext_vector_type(8))) float v8f;
typedef __attribute__((ext_vector_type(4))) unsigned int v4u;
typedef __attribute__((ext_vector_type(8))) int v8i;
typedef __attribute__((ext_vector_type(4))) int v4i;

// ---- k1 dynamic-LDS layout (float offsets) ----
// wl4:   [5][768]  padded weights, row4 = zeros, kw slot3 = zero   (3840)
// xs :   [64][3][68] input row tile, halo cols zero-padded         (13056)
// attn_s:[4][64]                                                    (256)
// ssum:  [64], ssq: [64]                                            (128)
#define K1_WL    0
#define K1_XS    3840
#define K1_ATTN  (K1_XS + 13056)
#define K1_SSUM  (K1_ATTN + 256)
#define K1_SSQ   (K1_SSUM + 64)
#define K1_LDS_FLOATS (K1_SSQ + 64)          // 17280 floats = 69120 B

// ---------------- K0: zero BN accumulators ----------------
__global__ void k0_init(float* __restrict__ ws) {
    int t = threadIdx.x;
    if (t < 128) ws[SUM_OFF + t] = 0.0f;   // covers SUM[64] + SQ[64]
}

// ---------------- K1: conv2d attention via fp32 WMMA + x2 + BN stats ----------------
// grid = N*H blocks, 128 threads (4 wave32).  Each block: one (n,h) row.
// Implicit GEMM: D[16 heads(pad) x 16 pixels] = A[16 x K] * B[K x 16]
// K padded to 64ch * 3kh * 4kw = 768; each WMMA K-chunk of 4 == one (c,kh) tap row.
__global__ __launch_bounds__(128) void k1_attn(const float* __restrict__ x,
                                               const float* __restrict__ attn_w,
                                               const float* __restrict__ attn_b,
                                               float* __restrict__ ws) {
    extern __shared__ float sm1[];
    float* __restrict__ wl4    = sm1 + K1_WL;
    float* __restrict__ xs     = sm1 + K1_XS;
    float* __restrict__ attn_s = sm1 + K1_ATTN;
    float* __restrict__ ssum   = sm1 + K1_SSUM;
    float* __restrict__ ssq    = sm1 + K1_SSQ;

    const int tid = threadIdx.x;
    const int n   = blockIdx.x / H_DIM;
    const int h   = blockIdx.x % H_DIM;
    const float* __restrict__ xb = x + (size_t)n * CHW;

    // prefetch next row of this image (global_prefetch_b8)
    if (h + 2 < H_DIM)
        __builtin_prefetch(xb + (tid & 63) * HW + (h + 2) * W_DIM, 0, 1);

    // padded weights -> LDS: wl4[row][c*12 + kh*4 + kw], kw==3 slot zero, row 4 zero
    for (int i = tid; i < 5 * 768; i += 128) {
        int row = i / 768;
        int rem = i - row * 768;
        int c   = rem / 12;
        int rr  = rem - c * 12;
        int kh  = rr >> 2;
        int kw  = rr & 3;
        float v = 0.0f;
        if (row < 4 && kw < 3) v = attn_w[row * 576 + c * 9 + kh * 3 + kw];
        wl4[i] = v;
    }

    // x row tile with halo: xs[c*204 + kh*68 + cc], x col = cc-1, zero pad
    for (int i = tid; i < 64 * 3 * 68; i += 128) {
        int c   = i / 204;
        int rem = i - c * 204;
        int r   = rem / 68;
        int cc  = rem - r * 68;
        int row = h + r - 1;
        int col = cc - 1;
        float v = 0.0f;
        if (row >= 0 && row < H_DIM && col >= 0 && col < W_DIM)
            v = xb[c * HW + row * W_DIM + col];
        xs[i] = v;
    }
    if (tid < 64) { ssum[tid] = 0.0f; ssq[tid] = 0.0f; }
    __syncthreads();

    // ---- WMMA main loop: 64ch * 3kh = 192 rank-4 updates, dual accumulators ----
    const int lane = tid & 31;
    const int wv   = tid >> 5;                 // wave -> pixel tile
    const int p    = (wv << 4) + (lane & 15);  // pixel column (0..63; >=56 padded)
    const int ksel = (lane & 16) ? 2 : 0;      // K sub-slot pair for this half-wave
    const int m    = lane & 15;                // A row (head, padded)
    const int wrow = (m < 4) ? m : 4;          // row 4 is the zero row

    const float* __restrict__ asrc = wl4 + wrow * 768 + ksel;  // 8B-aligned pairs
    const float* __restrict__ bsrc = xs + p + ksel;

    v8f acc0 = {0.f,0.f,0.f,0.f,0.f,0.f,0.f,0.f};
    v8f acc1 = {0.f,0.f,0.f,0.f,0.f,0.f,0.f,0.f};
    for (int c = 0; c < 64; c += 2) {
        const float* a0p = asrc + c * 12;
        const float* b0p = bsrc + c * 204;
#pragma unroll
        for (int kh = 0; kh < 3; ++kh) {
            v2f a0 = *(const v2f*)(a0p + kh * 4);                 // ds_load_b64
            v2f b0 = { b0p[kh * 68], b0p[kh * 68 + 1] };
            acc0 = __builtin_amdgcn_wmma_f32_16x16x4_f32(false, a0, false, b0,
                                                         (short)0, acc0, false, false);
            v2f a1 = *(const v2f*)(a0p + 12 + kh * 4);
            v2f b1 = { b0p[204 + kh * 68], b0p[204 + kh * 68 + 1] };
            acc1 = __builtin_amdgcn_wmma_f32_16x16x4_f32(false, a1, false, b1,
                                                         (short)0, acc1, false, false);
        }
    }
    v8f acc = acc0 + acc1;

    // heads 0..3 live in VGPR0..3 of lanes 0..15 (C/D layout)
    if (lane < 16) {
#pragma unroll
        for (int j = 0; j < 4; ++j) {
            float v = acc[j] + attn_b[j];
            v = fmaxf(v, 0.0f);
            v = 1.0f / (1.0f + __expf(-v));
            attn_s[j * 64 + p] = v;
        }
    }
    __syncthreads();

    float* __restrict__ x2 = ws + X2_OFF;

    // coalesced x2 stores (consecutive tid -> consecutive w)
    {
        int w  = tid & 63;
        int cb = (tid >> 6) * 32;
        if (w < W_DIM) {
            for (int cc = 0; cc < 32; ++cc) {
                int c = cb + cc;
                float v = xs[c * 204 + 68 + (w + 1)] * attn_s[(c >> 4) * 64 + w];
                x2[(size_t)(n * C_CH + c) * HW + h * W_DIM + w] = v;
            }
        }
    }
    // BN partial stats: channel-major per thread, then LDS reduce -> global atomics
    {
        int c = tid & 63;
        float s = 0.0f, q = 0.0f;
        for (int w = (tid >> 6); w < W_DIM; w += 2) {
            float v = xs[c * 204 + 68 + (w + 1)] * attn_s[(c >> 4) * 64 + w];
            s += v; q += v * v;
        }
        atomicAdd(&ssum[c], s);
        atomicAdd(&ssq[c], q);
    }
    __syncthreads();
    if (tid < 64) {
        atomicAdd(ws + SUM_OFF + tid, ssum[tid]);
        atomicAdd(ws + SQ_OFF + tid, ssq[tid]);
    }
}

// ---------------- K2: finalize BN scale/shift ----------------
__global__ void k2_stats(const float* __restrict__ gamma,
                         const float* __restrict__ beta,
                         float* __restrict__ ws) {
    int c = threadIdx.x;
    if (c < 64) {
        const float inv_cnt = 1.0f / (float)(N_IMG * HW);
        float mean = ws[SUM_OFF + c] * inv_cnt;
        float var  = ws[SQ_OFF + c] * inv_cnt - mean * mean;
        float sc   = gamma[c] * rsqrtf(var + EPS_BN);
        ws[SCALE_OFF + c] = sc;
        ws[SHIFT_OFF + c] = beta[c] - mean * sc;
    }
}

// ---------------- K3: grouped conv3d (3x3x3) + tanh -> gates ----------------
// grid = B*2*H blocks, 256 threads; block handles one (b, group, h) row across all T.
// dynamic LDS: wg[864] (staged via Tensor Data Mover) + xt[32][8][3][58]
__global__ __launch_bounds__(256) void k3_gate(const float* __restrict__ w3,
                                               const float* __restrict__ b3,
                                               float* __restrict__ ws) {
    extern __shared__ float sm3[];
    float* __restrict__ wg = sm3;          // 864 floats, at LDS offset 0 of dyn region
    float* __restrict__ xt = sm3 + 864;    // 32*8*3*58 = 44544 floats

    const int tid = threadIdx.x;
    const int h = blockIdx.x % H_DIM;
    const int g = (blockIdx.x / H_DIM) & 1;
    const int b = blockIdx.x / (2 * H_DIM);

    const float* __restrict__ x2    = ws + X2_OFF;
    const float* __restrict__ scale = ws + SCALE_OFF;
    const float* __restrict__ shift = ws + SHIFT_OFF;

#if __has_builtin(__builtin_amdgcn_tensor_load_to_lds) && __has_builtin(__builtin_amdgcn_s_wait_tensorcnt)
    // ---- TDM: DMA the 864 contiguous weight floats for this group into LDS ----
    if (tid < 32) {   // single wave issues the descriptor (EXEC-independent DMA)
        unsigned long long ga = (unsigned long long)(const void*)(w3 + g * 864);
        unsigned int lds_off = __builtin_amdgcn_groupstaticsize();   // dyn LDS base
        // D# group0: count=1, lds_addr, global_addr[56:0], type=2
        v4u d0 = { 1u, lds_off, (unsigned int)ga,
                   (unsigned int)((ga >> 32) & 0x1FFFFFFu) | (2u << 30) };
        // D# group1: data_size=4B; tensor_dim0=864; tensor_dim1=1; tile_dim0=864;
        //            tile_dim1/2 unused; tensor_dim0_stride=864
        v8i d1 = { (int)(2u << 16),
                   (int)(864u << 16),
                   (int)((864u >> 16) | (1u << 16)),
                   (int)(864u << 16),
                   0,
                   864,
                   0, 0 };
        v4i d2 = { 0, 0, 0, 0 };
        v4i d3 = { 0, 0, 0, 0 };
        v8i d4 = { 0, 0, 0, 0, 0, 0, 0, 0 };   // 6-arg toolchain: extra group
        __builtin_amdgcn_tensor_load_to_lds(d0, d1, d2, d3, d4, 0);
    }
#else
    for (int i = tid; i < 864; i += 256) wg[i] = w3[g * 864 + i];
#endif

    // stage relu(bn(x2)) tile while the TDM runs: index ((c*8+t)*3+r)*58 + wcol
    for (int i = tid; i < 44544; i += 256) {
        int wcol = i % 58;
        int j    = i / 58;
        int r    = j % 3;  j /= 3;
        int t    = j & 7;
        int c    = j >> 3;
        int row  = h + r - 1;
        int col  = wcol - 1;
        float v = 0.0f;
        if (row >= 0 && row < H_DIM && col >= 0 && col < W_DIM) {
            int cg = g * 32 + c;
            float xv = x2[(size_t)((b * T_DIM + t) * C_CH + cg) * HW + row * W_DIM + col];
            v = fmaxf(fmaf(xv, scale[cg], shift[cg]), 0.0f);
        }
        xt[i] = v;
    }

#if __has_builtin(__builtin_amdgcn_tensor_load_to_lds) && __has_builtin(__builtin_amdgcn_s_wait_tensorcnt)
    if (tid < 32) __builtin_amdgcn_s_wait_tensorcnt(0);
#endif
    __syncthreads();

    const float bias = b3[g];
    float* __restrict__ gate = ws + GATE_OFF;

    for (int o = tid; o < T_DIM * W_DIM; o += 256) {
        int t = o / W_DIM;
        int w = o - t * W_DIM;
        float acc = bias;
        for (int kt = 0; kt < 3; ++kt) {
            int tt = t + kt - 1;
            if (tt < 0 || tt >= T_DIM) continue;   // SAME zero pad in T
            for (int c = 0; c < 32; ++c) {
                const float* xp = &xt[((c * 8 + tt) * 3) * 58];
                const float* wp = &wg[c * 27 + kt * 9];
#pragma unroll
                for (int kh = 0; kh < 3; ++kh)
#pragma unroll
                    for (int kw = 0; kw < 3; ++kw)
                        acc = fmaf(xp[kh * 58 + w + kw], wp[kh * 3 + kw], acc);
            }
        }
        gate[(size_t)((b * 2 + g) * T_DIM + t) * HW + h * W_DIM + w] = tanhf(acc);
    }
}

// ---------------- K4: temporal-shift gating + interleave, float4 vectorized ----------------
__global__ __launch_bounds__(256) void k4_out(float* __restrict__ out,
                                              const float* __restrict__ ws) {
    int q = blockIdx.x * 256 + threadIdx.x;      // quad index, w%4==0 within rows
    if (q >= TOTAL / 4) return;
    int idx = q * 4;

    const float* __restrict__ x2   = ws + X2_OFF;
    const float* __restrict__ gate = ws + GATE_OFF;

    int w = idx % W_DIM;
    int h = (idx / W_DIM) % H_DIM;
    int c = (idx / HW) & 63;
    int n = idx / CHW;
    int b = n >> 3;          // n / T
    int t = n & 7;           // n % T

    int grp = c >> 5;
    int cl  = c & 31;
    int src_c = grp * 32 + ((cl & 1) << 4) + (cl >> 1);   // interleave inverse
    int hw = h * W_DIM + w;

    size_t xi = (size_t)((b * T_DIM + t) * C_CH + src_c) * HW + hw;
    size_t gi = (size_t)((b * 2 + grp) * T_DIM + t) * HW + hw;

    float4 xcur = *(const float4*)(x2 + xi);     // global_load_b128
    float4 gcur = *(const float4*)(gate + gi);

    float4 o;
    int tn = (grp == 0) ? (t + 1) : (t - 1);     // group0 shifts fwd, group1 bwd
    if (tn >= 0 && tn < T_DIM) {
        size_t d = (size_t)(tn - t);
        float4 xn = *(const float4*)(x2 + xi + d * C_CH * HW);
        float4 gn = *(const float4*)(gate + gi + d * HW);
        o.x = gn.x * xn.x + (1.0f - gcur.x) * xcur.x;
        o.y = gn.y * xn.y + (1.0f - gcur.y) * xcur.y;
        o.z = gn.z * xn.z + (1.0f - gcur.z) * xcur.z;
        o.w = gn.w * xn.w + (1.0f - gcur.w) * xcur.w;
    } else {
        o.x = (1.0f - gcur.x) * xcur.x;
        o.y = (1.0f - gcur.y) * xcur.y;
        o.z = (1.0f - gcur.z) * xcur.z;
        o.w = (1.0f - gcur.w) * xcur.w;
    }
    *(float4*)(out + idx) = o;                   // global_store_b128
}

// ---------------- launch ----------------
extern "C" void kernel_launch(void* const* d_in, const int* in_sizes, int n_in,
                              void* d_out, int out_size, void* d_ws, size_t ws_size,
                              hipStream_t stream) {
    (void)in_sizes; (void)n_in; (void)out_size; (void)ws_size;

    const float* x        = (const float*)d_in[0];
    const float* attn_w   = (const float*)d_in[1];
    const float* attn_b   = (const float*)d_in[2];
    const float* bn_gamma = (const float*)d_in[3];
    const float* bn_beta  = (const float*)d_in[4];
    const float* conv3d_w = (const float*)d_in[5];
    const float* conv3d_b = (const float*)d_in[6];
    float* ws  = (float*)d_ws;
    float* out = (float*)d_out;

    k0_init<<<1, 128, 0, stream>>>(ws);
    k1_attn<<<N_IMG * H_DIM, 128, K1_LDS_FLOATS * sizeof(float), stream>>>(x, attn_w, attn_b, ws);
    k2_stats<<<1, 64, 0, stream>>>(bn_gamma, bn_beta, ws);
    k3_gate<<<B_DIM * 2 * H_DIM, 256, (864 + 44544) * sizeof(float), stream>>>(conv3d_w, conv3d_b, ws);
    k4_out<<<TOTAL / 4 / 256, 256, 0, stream>>>(out, ws);
}